// AnchorTargetLayer_46385646797680
// MI455X (gfx1250) — compile-verified
//
#include <hip/hip_runtime.h>
#include <hip/hip_bf16.h>
#include <math.h>
#include <stdint.h>

// AnchorTargetLayer for MI455X (gfx1250, wave32).
// Memory/latency-bound (~25MB @ 23.3TB/s); no GEMM structure -> no WMMA.
// CDNA5 paths: global_load_async_to_lds_b128 + s_wait_asynccnt to stage GT
// boxes (512B) and anchor tiles (4KB) into LDS, ds_max_u64 LDS atomics.

#define B_    16
#define G_    32
#define NTOT  65472
#define NLEV  5
#define HI_   0.5f
#define LO_   0.4f

typedef unsigned long long ull;

__constant__ int d_lvl_off[NLEV + 1] = {0, 49152, 61440, 64512, 65280, 65472};

__device__ __forceinline__ int level_of(int a) {
    if (a < 49152) return 0;
    if (a < 61440) return 1;
    if (a < 64512) return 2;
    if (a < 65280) return 3;
    return 4;
}

// One lane -> one float4 moved by the CDNA5 async engine into LDS.
__device__ __forceinline__ void async_b128(unsigned lds_off, const void* gsrc) {
    asm volatile("global_load_async_to_lds_b128 %0, %1, off"
                 :: "v"(lds_off), "v"(gsrc) : "memory");
}
__device__ __forceinline__ void wait_async0() {
    asm volatile("s_wait_asynccnt 0" ::: "memory");
}

__global__ void kInit(int* __restrict__ forced,
                      ull* __restrict__ gt_best) {
    int i = blockIdx.x * 256 + threadIdx.x;
    if (i < B_ * NTOT) forced[i] = -1;
    if (i < NLEV * B_ * G_) gt_best[i] = 0ull;
}

// Per (level, b, g): max IoU over the level's anchors + argmax anchor index.
// Key = {iou_bits[31:0], ~anchor_idx}: larger IoU wins; ties -> smaller anchor
// index (matches jnp.argmax first-max). IoU >= 0 so float bits are monotonic.
// Transposed reduction: thread t owns GT g=t&31 and anchor chunk c=t>>5 (32
// anchors from LDS, wave-uniform broadcast reads); private running max-key,
// then one LDS ds_max_u64 per thread, then 32 global atomics per block.
__global__ void __launch_bounds__(256)
kBestPerGT(const float* __restrict__ bb, const float* __restrict__ anchors,
           ull* __restrict__ gt_best) {
    __shared__ float  shGT[G_ * 4];
    __shared__ float4 shA[256];
    __shared__ ull    shBest[G_];
    const int b   = blockIdx.y;
    const int tid = threadIdx.x;
    const int a0  = blockIdx.x * 256;

    // Async stage: this block's 256 anchors (4KB) + this batch's GT boxes.
    {
        int al = a0 + tid; if (al >= NTOT) al = NTOT - 1;   // clamp, masked later
        async_b128((unsigned)(uintptr_t)(&shA[tid]), anchors + (size_t)al * 4);
    }
    if (tid < G_) {
        async_b128((unsigned)(uintptr_t)(&shGT[tid * 4]), bb + (size_t)(b * G_ + tid) * 4);
        shBest[tid] = 0ull;
    }
    wait_async0();          // each wave waits for its own ASYNCcnt
    __syncthreads();

    const int li = level_of(a0);                    // uniform: bounds 256-aligned
    const unsigned base_aidx = (unsigned)(a0 - d_lvl_off[li]);

    const int g  = tid & 31;
    const int c0 = (tid >> 5) * 32;
    const float gx1 = shGT[g*4+0], gy1 = shGT[g*4+1];
    const float gx2 = shGT[g*4+2], gy2 = shGT[g*4+3];
    const float ag  = (gx2 - gx1) * (gy2 - gy1);

    ull best = 0ull;
    #pragma unroll 8
    for (int j = 0; j < 32; ++j) {
        const int idx = c0 + j;
        float4 A = shA[idx];                         // broadcast (uniform in wave)
        float aa = (A.z - A.x) * (A.w - A.y);
        float x1 = fmaxf(A.x, gx1), y1 = fmaxf(A.y, gy1);
        float x2 = fminf(A.z, gx2), y2 = fminf(A.w, gy2);
        float inter = fmaxf(x2 - x1, 0.f) * fmaxf(y2 - y1, 0.f);
        float v = inter / (aa + ag - inter + 1e-9f);
        if (a0 + idx < NTOT) {
            ull key = ((ull)__float_as_uint(v) << 32) | (ull)(~(base_aidx + (unsigned)idx));
            best = (key > best) ? key : best;
        }
    }
    atomicMax(&shBest[g], best);                     // ds_max_u64, 8-way
    __syncthreads();
    if (tid < G_)
        atomicMax(&gt_best[((size_t)li * B_ + b) * G_ + tid], shBest[tid]);
}

// Per (b,g): best level (first-max over levels), recover its argmax anchor,
// scatter g into forced[] (largest g wins duplicate targets).
__global__ void kForce(const ull* __restrict__ gt_best,
                       int* __restrict__ forced) {
    int b = blockIdx.x;
    int g = threadIdx.x;
    float best = -1.0f; int bli = 0; unsigned baidx = 0;
    #pragma unroll
    for (int li = 0; li < NLEV; ++li) {
        ull k = gt_best[((size_t)li * B_ + b) * G_ + g];
        float v = __uint_as_float((unsigned)(k >> 32));
        if (v > best) { best = v; bli = li; baidx = ~(unsigned)(k & 0xFFFFFFFFull); }
    }
    int ga = d_lvl_off[bli] + (int)baidx;
    atomicMax(&forced[(size_t)b * NTOT + ga], g);
}

// Output pass: recompute the 32 IoUs per anchor (cheaper than spilling 8MB of
// max/arg to HBM), apply forced assignments, emit labels (as float) + regs.
__global__ void __launch_bounds__(256)
kOutput(const float* __restrict__ bb, const int* __restrict__ ids,
        const float* __restrict__ anchors, const int* __restrict__ forced,
        float* __restrict__ out) {
    __shared__ float shGT[G_ * 4];
    __shared__ int   shID[G_];
    const int b   = blockIdx.y;
    const int tid = threadIdx.x;
    const int a   = blockIdx.x * 256 + tid;

    if (tid < G_) {
        async_b128((unsigned)(uintptr_t)(&shGT[tid * 4]), bb + (size_t)(b * G_ + tid) * 4);
        wait_async0();
        shID[tid] = ids[b * G_ + tid];
    }
    __syncthreads();
    if (a >= NTOT) return;

    float4 A  = *(const float4*)(anchors + (size_t)a * 4);   // coalesced b128
    float  aa = (A.z - A.x) * (A.w - A.y);

    float maxv = -1.0f; int argg = 0;
    #pragma unroll 4
    for (int g = 0; g < G_; ++g) {
        float gx1 = shGT[g*4+0], gy1 = shGT[g*4+1];
        float gx2 = shGT[g*4+2], gy2 = shGT[g*4+3];
        float x1 = fmaxf(A.x, gx1), y1 = fmaxf(A.y, gy1);
        float x2 = fminf(A.z, gx2), y2 = fminf(A.w, gy2);
        float inter = fmaxf(x2 - x1, 0.f) * fmaxf(y2 - y1, 0.f);
        float ag = (gx2 - gx1) * (gy2 - gy1);
        float v  = inter / (aa + ag - inter + 1e-9f);
        if (v > maxv) { maxv = v; argg = g; }        // strict '>' = first max
    }

    const int  f    = forced[(size_t)b * NTOT + a];
    const bool pos  = (f >= 0) || (maxv >= HI_);     // forced overrides
    const int  gsel = (f >= 0) ? f : argg;

    float label;
    if (pos)              label = (float)shID[gsel];
    else if (maxv >= LO_) label = -1.0f;             // ignore band [LO, HI)
    else                  label = 0.0f;

    const size_t oL = (size_t)b * NTOT + a;
    out[oL] = label;

    float4 R = make_float4(0.f, 0.f, 0.f, 0.f);
    if (pos) {
        float gx1 = shGT[gsel*4+0], gy1 = shGT[gsel*4+1];
        float gx2 = shGT[gsel*4+2], gy2 = shGT[gsel*4+3];
        float aw = A.z - A.x, ah = A.w - A.y;
        float ax = A.x + 0.5f * aw, ay = A.y + 0.5f * ah;
        float gw = gx2 - gx1, gh = gy2 - gy1;
        float gx = gx1 + 0.5f * gw, gy = gy1 + 0.5f * gh;
        R = make_float4((gx - ax) / aw, (gy - ay) / ah, logf(gw / aw), logf(gh / ah));
    }
    ((float4*)(out + (size_t)B_ * NTOT))[oL] = R;    // 16B-aligned
}

extern "C" void kernel_launch(void* const* d_in, const int* in_sizes, int n_in,
                              void* d_out, int out_size, void* d_ws, size_t ws_size,
                              hipStream_t stream) {
    const float* bb      = (const float*)d_in[0];   // (16,32,4) f32
    const int*   ids     = (const int*)  d_in[1];   // (16,32)   i32
    const float* anchors = (const float*)d_in[2];   // (65472,4) f32
    float*       out     = (float*)d_out;           // labels then regs, flat

    ull* gt_best = (ull*)d_ws;                                      // 20,480 B
    int* forced  = (int*)((char*)d_ws + (size_t)NLEV * B_ * G_ * 8); // 4,190,208 B

    const int nblk = (NTOT + 255) / 256;            // 256 level-aligned blocks

    kInit     <<<dim3((B_ * NTOT + 255) / 256), 256, 0, stream>>>(forced, gt_best);
    kBestPerGT<<<dim3(nblk, B_),                256, 0, stream>>>(bb, anchors, gt_best);
    kForce    <<<dim3(B_),                      G_,  0, stream>>>(gt_best, forced);
    kOutput   <<<dim3(nblk, B_),                256, 0, stream>>>(bb, ids, anchors, forced, out);
}